// DenseKANRBF_47940424958484
// MI455X (gfx1250) — compile-verified
//
#include <hip/hip_runtime.h>
#include <math.h>

typedef __attribute__((ext_vector_type(2))) float v2f;
typedef __attribute__((ext_vector_type(8))) float v8f;
typedef __attribute__((ext_vector_type(4))) unsigned int u32x4;
typedef __attribute__((ext_vector_type(8))) int i32x8;
typedef __attribute__((ext_vector_type(4))) int i32x4;

#define Bsz  1024
#define Fsz  512
#define Gsz  8
#define Usz  512
#define Hsz  2048
#define KPHI (Fsz * Gsz)        // 4096
#define KCAT (KPHI + Hsz)       // 6144

#define AS_STRIDE 18            // 16 + 2 pad dwords (TDM pad: interval 16dw, amount 2dw)
#define BS_STRIDE 66            // 64 + 2 pad dwords (TDM pad: interval 64dw, amount 2dw)
#define A_BUF_BYTES (64 * AS_STRIDE * 4)
#define B_BUF_BYTES (16 * BS_STRIDE * 4)

// ---------------------------------------------------------------------------
// Kernel 1: RBF basis expansion. Phi[b][f*8+g] = exp(-(x[b][f]-c_g)^2).
// ---------------------------------------------------------------------------
__global__ __launch_bounds__(256) void phi_kernel(const float* __restrict__ x,
                                                  float* __restrict__ phi) {
    const int idx = blockIdx.x * 256 + threadIdx.x;     // 0 .. B*F-1 (exact grid)
    const float xv = x[idx];
    float* __restrict__ op = phi + (size_t)idx * Gsz;
#pragma unroll
    for (int g = 0; g < Gsz; ++g) {
        const float c = -1.0f + (2.0f / 7.0f) * (float)g;
        const float d = xv - c;
        op[g] = expf(-d * d);
    }
}

// ---------------------------------------------------------------------------
// V_WMMA_F32_16X16X4_F32 helper (wave32).
// A 16x4:  lanes 0-15 -> M=lane, K={0,1}; lanes 16-31 -> M=lane-16, K={2,3}
// B 4x16:  VGPR v: lanes 0-15 -> K=v, N=lane; lanes 16-31 -> K=2+v, N=lane-16
// C/D 16x16: VGPR v: lanes 0-15 -> M=v, N=lane; lanes 16-31 -> M=8+v, N=lane-16
// ---------------------------------------------------------------------------
__device__ __forceinline__ v8f wmma4(v2f a, v2f b, v8f c) {
    return __builtin_amdgcn_wmma_f32_16x16x4_f32(
        /*neg_a=*/false, a, /*neg_b=*/false, b,
        /*c_mod=*/(short)0, c, /*reuse_a=*/false, /*reuse_b=*/false);
}

// ---------------------------------------------------------------------------
// TDM: 2-D tile load Global -> LDS, with LDS padding so the deposited layout
// matches the bank-conflict-free strides used by the compute code.
// Descriptor bit packing per CDNA5 ISA §8.3/§8.4 (D# group0 / group1).
//   group0: [1:0]=count=1, [63:32]=lds_addr, [120:64]=global_addr, [127:126]=2
//   group1: data_size=4B, pad_enable, pad_interval/amount codes,
//           tensor_dim0/1, tile_dim0/1 (tile_dim2=0 -> 2D), dim0 stride.
// 6-arg builtin (clang-23 / therock-10.0): extra zero i32x8 group before cpol.
// ---------------------------------------------------------------------------
__device__ __forceinline__ void tdm_load_2d(unsigned lds_off, const float* gptr,
                                            unsigned tensor_d0, unsigned tensor_d1,
                                            unsigned tile_d0, unsigned tile_d1,
                                            unsigned stride_d0,
                                            unsigned pad_int_code, unsigned pad_amt_code) {
    const unsigned long long ga = (unsigned long long)(size_t)gptr;
    u32x4 g0;
    g0.x = 1u;                                            // count=1, user descriptor
    g0.y = lds_off;                                       // LDS byte address
    g0.z = (unsigned)(ga & 0xFFFFFFFFull);                // global_addr[31:0]
    g0.w = (unsigned)((ga >> 32) & 0x01FFFFFFull) | (2u << 30);   // [56:32] | type=2
    i32x8 g1;
    g1[0] = (int)((2u << 16) |                            // data_size = 4 bytes
                  (1u << 20) |                            // pad_enable
                  (pad_int_code << 22) | (pad_amt_code << 25));
    g1[1] = (int)((tensor_d0 & 0xFFFFu) << 16);           // td0[15:0] (barrier addr = 0)
    g1[2] = (int)(((tensor_d0 >> 16) & 0xFFFFu) | ((tensor_d1 & 0xFFFFu) << 16));
    g1[3] = (int)(((tensor_d1 >> 16) & 0xFFFFu) | ((tile_d0 & 0xFFFFu) << 16));
    g1[4] = (int)(tile_d1 & 0xFFFFu);                     // tile_dim2 = 0 (2D)
    g1[5] = (int)stride_d0;                               // dim0 stride [31:0]
    g1[6] = 0;                                            // stride[47:32]=0, dim1 stride=0
    g1[7] = 0;
    const i32x4 z4 = {0, 0, 0, 0};
    const i32x8 z8 = {0, 0, 0, 0, 0, 0, 0, 0};
    __builtin_amdgcn_tensor_load_to_lds(g0, g1, z4, z4, z8, 0);
}

// ---------------------------------------------------------------------------
// Kernel 2: H = gelu_exact(x @ W1 + b1).  (1024x512)x(512x2048).
// 64x64 tile/block; 8 waves in 4(M)x2(N); wave tile 16x32 (2 f32 accums).
// Double-buffered LDS tiles filled by the Tensor Data Mover (wave 0 issues).
// ---------------------------------------------------------------------------
__global__ __launch_bounds__(256) void h_gemm(const float* __restrict__ x,
                                              const float* __restrict__ w1,
                                              const float* __restrict__ b1,
                                              float* __restrict__ hout) {
    __shared__ float As[2][64][AS_STRIDE];
    __shared__ float Bs[2][16][BS_STRIDE];

    const int colBase = blockIdx.x * 64;
    const int rowBase = blockIdx.y * 64;
    const int t    = threadIdx.x;
    const int lane = t & 31;
    const int wave = t >> 5;
    const int wm   = wave & 3;
    const int wn   = wave >> 2;
    const bool hi  = lane >= 16;
    const int ml   = lane & 15;
    const bool issuer = (wave == 0);                      // wave-uniform branch

    const unsigned ldsA = (unsigned)(size_t)&As[0][0][0]; // LDS aperture: low 32 bits
    const unsigned ldsB = (unsigned)(size_t)&Bs[0][0][0];

    v8f acc0 = {}; v8f acc1 = {};
    const int NT = Fsz / 16;                              // 32 K-tiles

    if (issuer) {
        tdm_load_2d(ldsA, x + (size_t)rowBase * Fsz, Fsz, Bsz, 16, 64, Fsz, 3, 1);
        tdm_load_2d(ldsB, w1 + (size_t)colBase, Hsz, Fsz, 64, 16, Hsz, 5, 1);
    }
    for (int i = 0; i < NT; ++i) {
        if (issuer) {
            if (i + 1 < NT) {
                const int kn = (i + 1) * 16;
                const int bn = (i + 1) & 1;
                tdm_load_2d(ldsA + bn * A_BUF_BYTES, x + (size_t)rowBase * Fsz + kn,
                            Fsz, Bsz, 16, 64, Fsz, 3, 1);
                tdm_load_2d(ldsB + bn * B_BUF_BYTES, w1 + (size_t)kn * Hsz + colBase,
                            Hsz, Fsz, 64, 16, Hsz, 5, 1);
                __builtin_amdgcn_s_wait_tensorcnt(2);     // tile i's 2 DMAs done
            } else {
                __builtin_amdgcn_s_wait_tensorcnt(0);
            }
        }
        __syncthreads();                                  // tile i visible to all waves
        const float (*A)[AS_STRIDE] = As[i & 1];
        const float (*Bt)[BS_STRIDE] = Bs[i & 1];
#pragma unroll
        for (int kk = 0; kk < 4; ++kk) {
            const int kb = kk * 4 + (hi ? 2 : 0);
            v2f a;   a.x   = A[wm * 16 + ml][kb];          a.y   = A[wm * 16 + ml][kb + 1];
            v2f b0;  b0.x  = Bt[kb][wn * 32 + ml];         b0.y  = Bt[kb + 1][wn * 32 + ml];
            v2f b1v; b1v.x = Bt[kb][wn * 32 + 16 + ml];    b1v.y = Bt[kb + 1][wn * 32 + 16 + ml];
            acc0 = wmma4(a, b0,  acc0);
            acc1 = wmma4(a, b1v, acc1);
        }
        __syncthreads();                                  // safe to overwrite buffer i&1
    }

    const int gc0 = colBase + wn * 32 + ml;
    const int gc1 = gc0 + 16;
    const float bb0 = b1[gc0], bb1 = b1[gc1];
    const int r0 = rowBase + wm * 16 + (hi ? 8 : 0);
#pragma unroll
    for (int v = 0; v < 8; ++v) {
        float v0 = acc0[v] + bb0;
        float v1 = acc1[v] + bb1;
        v0 = 0.5f * v0 * (1.0f + erff(v0 * 0.7071067811865475f));   // exact gelu
        v1 = 0.5f * v1 * (1.0f + erff(v1 * 0.7071067811865475f));
        hout[(size_t)(r0 + v) * Hsz + gc0] = v0;
        hout[(size_t)(r0 + v) * Hsz + gc1] = v1;
    }
}

// ---------------------------------------------------------------------------
// Kernel 3: fused out = [Phi | H] @ [basis_kernel ; W2] + b2 + bias.
// K = 4096 (Phi/basis_kernel) then 2048 (H/W2); region uniform per K-tile.
// ---------------------------------------------------------------------------
__device__ __forceinline__ void out_issue(int kt, int buf, int rowBase, int colBase,
                                          const float* phi, const float* hbuf,
                                          const float* bk, const float* w2,
                                          unsigned ldsA, unsigned ldsB) {
    const float* asrc; const float* bsrc; unsigned astride; unsigned kdim; int k0;
    if (kt < KPHI) { asrc = phi;  astride = KPHI; kdim = KPHI; k0 = kt;        bsrc = bk; }
    else           { asrc = hbuf; astride = Hsz;  kdim = Hsz;  k0 = kt - KPHI; bsrc = w2; }
    tdm_load_2d(ldsA + buf * A_BUF_BYTES, asrc + (size_t)rowBase * astride + k0,
                astride, Bsz, 16, 64, astride, 3, 1);
    tdm_load_2d(ldsB + buf * B_BUF_BYTES, bsrc + (size_t)k0 * Usz + colBase,
                Usz, kdim, 64, 16, Usz, 5, 1);
}

__global__ __launch_bounds__(256) void out_gemm(const float* __restrict__ phi,
                                                const float* __restrict__ hbuf,
                                                const float* __restrict__ bk,
                                                const float* __restrict__ w2,
                                                const float* __restrict__ b2,
                                                const float* __restrict__ bias,
                                                float* __restrict__ out) {
    __shared__ float As[2][64][AS_STRIDE];
    __shared__ float Bs[2][16][BS_STRIDE];

    const int colBase = blockIdx.x * 64;
    const int rowBase = blockIdx.y * 64;
    const int t    = threadIdx.x;
    const int lane = t & 31;
    const int wave = t >> 5;
    const int wm   = wave & 3;
    const int wn   = wave >> 2;
    const bool hi  = lane >= 16;
    const int ml   = lane & 15;
    const bool issuer = (wave == 0);

    const unsigned ldsA = (unsigned)(size_t)&As[0][0][0];
    const unsigned ldsB = (unsigned)(size_t)&Bs[0][0][0];

    v8f acc0 = {}; v8f acc1 = {};
    const int NT = KCAT / 16;                             // 384 K-tiles

    if (issuer) out_issue(0, 0, rowBase, colBase, phi, hbuf, bk, w2, ldsA, ldsB);
    for (int i = 0; i < NT; ++i) {
        if (issuer) {
            if (i + 1 < NT) {
                out_issue((i + 1) * 16, (i + 1) & 1, rowBase, colBase,
                          phi, hbuf, bk, w2, ldsA, ldsB);
                __builtin_amdgcn_s_wait_tensorcnt(2);
            } else {
                __builtin_amdgcn_s_wait_tensorcnt(0);
            }
        }
        __syncthreads();
        const float (*A)[AS_STRIDE] = As[i & 1];
        const float (*Bt)[BS_STRIDE] = Bs[i & 1];
#pragma unroll
        for (int kk = 0; kk < 4; ++kk) {
            const int kb = kk * 4 + (hi ? 2 : 0);
            v2f a;   a.x   = A[wm * 16 + ml][kb];          a.y   = A[wm * 16 + ml][kb + 1];
            v2f b0;  b0.x  = Bt[kb][wn * 32 + ml];         b0.y  = Bt[kb + 1][wn * 32 + ml];
            v2f b1v; b1v.x = Bt[kb][wn * 32 + 16 + ml];    b1v.y = Bt[kb + 1][wn * 32 + 16 + ml];
            acc0 = wmma4(a, b0,  acc0);
            acc1 = wmma4(a, b1v, acc1);
        }
        __syncthreads();
    }

    const int gc0 = colBase + wn * 32 + ml;
    const int gc1 = gc0 + 16;
    const float add0 = b2[gc0] + bias[gc0];
    const float add1 = b2[gc1] + bias[gc1];
    const int r0 = rowBase + wm * 16 + (hi ? 8 : 0);
#pragma unroll
    for (int v = 0; v < 8; ++v) {
        out[(size_t)(r0 + v) * Usz + gc0] = acc0[v] + add0;
        out[(size_t)(r0 + v) * Usz + gc1] = acc1[v] + add1;
    }
}

// ---------------------------------------------------------------------------
// Launch: phi -> H -> fused output GEMM. Workspace: Phi 16MB + H 8MB.
// Inputs: x, basis_kernel, mlp_w1, mlp_b1, mlp_w2, mlp_b2, bias (all fp32).
// ---------------------------------------------------------------------------
extern "C" void kernel_launch(void* const* d_in, const int* in_sizes, int n_in,
                              void* d_out, int out_size, void* d_ws, size_t ws_size,
                              hipStream_t stream) {
    const float* x    = (const float*)d_in[0];
    const float* bk   = (const float*)d_in[1];
    const float* w1   = (const float*)d_in[2];
    const float* b1   = (const float*)d_in[3];
    const float* w2   = (const float*)d_in[4];
    const float* b2   = (const float*)d_in[5];
    const float* bias = (const float*)d_in[6];
    float* out = (float*)d_out;

    float* phi  = (float*)d_ws;                       // B x 4096
    float* hbuf = phi + (size_t)Bsz * KPHI;           // B x 2048

    phi_kernel<<<(Bsz * Fsz) / 256, 256, 0, stream>>>(x, phi);
    h_gemm<<<dim3(Hsz / 64, Bsz / 64), 256, 0, stream>>>(x, w1, b1, hbuf);
    out_gemm<<<dim3(Usz / 64, Bsz / 64), 256, 0, stream>>>(phi, hbuf, bk, w2, b2, bias, out);
}